// REANN_60120952209686
// MI455X (gfx1250) — compile-verified
//
#include <hip/hip_runtime.h>
#include <hip/hip_bf16.h>

#define NS    4
#define NMAX  16
#define LMAX  2
#define LOOPN 2
#define NO    4
#define HID   128
#define NTA   50000
#define NN    1600000
#define PI_F  3.14159265358979323846f
#define RCUT_F 6.0f

typedef float v2f __attribute__((ext_vector_type(2)));
typedef float v8f __attribute__((ext_vector_type(8)));

// ---------------------------------------------------------------------------
// Csn[t][m] = species_params[symbols[t]][m]
// ---------------------------------------------------------------------------
__global__ void k_init_csn(const float* __restrict__ sp,
                           const int* __restrict__ symbols,
                           float* __restrict__ Csn) {
  int i = blockIdx.x * blockDim.x + threadIdx.x;
  if (i >= NTA * NMAX) return;
  int t = i >> 4;
  int m = i & 15;
  Csn[i] = sp[symbols[t] * NMAX + m];
}

__global__ void k_zero(float* __restrict__ p, int n) {
  int i = blockIdx.x * blockDim.x + threadIdx.x;
  if (i < n) p[i] = 0.0f;
}

// ---------------------------------------------------------------------------
// Pair scatter: thread = (pair n, radial m). 4 atomics per thread (o=0..3).
// bnl layout: [t][o][m], m contiguous. iidx is random -> L2-resident f32
// atomics are the right path (bnl = 12.8MB << 192MB L2).
// ---------------------------------------------------------------------------
__global__ void k_pairs(const float* __restrict__ disp,
                        const float* __restrict__ dist,
                        const float* __restrict__ alpha,
                        const float* __restrict__ rs,
                        const float* __restrict__ Csn,
                        const int* __restrict__ iidx,
                        const int* __restrict__ jidx,
                        const int* __restrict__ jsym,
                        float* __restrict__ bnl) {
  int gid = blockIdx.x * blockDim.x + threadIdx.x;   // < NN*NMAX = 25.6M
  if (gid >= NN * NMAX) return;
  int n = gid >> 4;
  int m = gid & 15;

  // forward-stream prefetch of the sequential pair arrays (global_prefetch_b8)
  if (n + 8192 < NN) {
    __builtin_prefetch(&dist[n + 8192], 0, 0);
    __builtin_prefetch(&disp[(n + 8192) * 3], 0, 0);
  }

  float d  = dist[n];
  int   js = jsym[n];
  float a  = alpha[js * NMAX + m];
  float r0 = rs[js * NMAX + m];
  float dr = d - r0;
  float rad = __expf(a * dr * dr);
  float cs  = __cosf(d * (PI_F / RCUT_F));
  float fc  = 0.25f * (cs + 1.0f) * (cs + 1.0f);
  float cj  = Csn[jidx[n] * NMAX + m];
  float base = fc * cj * rad;
  float dx = disp[n * 3 + 0];
  float dy = disp[n * 3 + 1];
  float dz = disp[n * 3 + 2];
  float* dst = bnl + (size_t)iidx[n] * (NO * NMAX) + m;
  atomicAdd(dst + 0 * NMAX, base);        // l=0 : ang = 1
  atomicAdd(dst + 1 * NMAX, base * dx);   // l=1 : ang = dx
  atomicAdd(dst + 2 * NMAX, base * dy);   // l=2 : ang = dy
  atomicAdd(dst + 3 * NMAX, base * dz);   // l=3 : ang = dz
}

// ---------------------------------------------------------------------------
// Density: one wave per 16-atom tile.
//   t[M][o] = sum_n bnl[t0+M][l][n] * W_l[n][o]  via 4 chained
//   V_WMMA_F32_16X16X4_F32 (K = NMAX split into 4 chunks of 4), then
//   rho[M][o] = sum_l t^2.
// A layout (32-bit 16x4): lanes0-15 K={0,1}, lanes16-31 K={2,3} -> float2 load.
// B layout (4x16): VGPR0 = row K0+2*half, VGPR1 = row K0+2*half+1, N = lane&15.
// D layout: VGPR r -> M = r + 8*(lane>>4), N = lane&15.
// oidx = [0,1,1,1] -> only two distinct W matrices; all 8 B register pairs
// are preloaded branchlessly (clamped address * lane mask) before the chain.
// ---------------------------------------------------------------------------
__global__ void k_density(const float* __restrict__ bnl,
                          const float* __restrict__ op,   // LMAX x NMAX x NO
                          float* __restrict__ rho) {
  int wave = threadIdx.x >> 5;
  int lane = threadIdx.x & 31;
  int t0 = (blockIdx.x * (blockDim.x >> 5) + wave) * 16;
  if (t0 >= NTA) return;            // wave-uniform: EXEC stays all-ones
  int half = lane >> 4;             // 0 or 1
  int lm   = lane & 15;

  float mask = (lm < NO) ? 1.0f : 0.0f;   // zero-pad B columns o >= NO
  int   oc   = (lm < NO) ? lm : (NO - 1); // clamped (always in-bounds) column

  // Preload B: 2 matrices x 4 K-chunks, no divergence.
  v2f B0[4], B1[4];
  #pragma unroll
  for (int c = 0; c < 4; ++c) {
    int kk = c * 4 + 2 * half;
    B0[c].x = op[(kk + 0) * NO + oc] * mask;
    B0[c].y = op[(kk + 1) * NO + oc] * mask;
    B1[c].x = op[NMAX * NO + (kk + 0) * NO + oc] * mask;
    B1[c].y = op[NMAX * NO + (kk + 1) * NO + oc] * mask;
  }

  v8f acc = {0.f, 0.f, 0.f, 0.f, 0.f, 0.f, 0.f, 0.f};

  #pragma unroll
  for (int l = 0; l < NO; ++l) {
    const float* arow = bnl + (size_t)(t0 + lm) * (NO * NMAX) + l * NMAX + 2 * half;
    v8f c = {0.f, 0.f, 0.f, 0.f, 0.f, 0.f, 0.f, 0.f};
    #pragma unroll
    for (int k0c = 0; k0c < 4; ++k0c) {
      v2f a = *(const v2f*)(arow + 4 * k0c);  // 8B-aligned (even float offset)
      v2f b = (l == 0) ? B0[k0c] : B1[k0c];
      c = __builtin_amdgcn_wmma_f32_16x16x4_f32(
              /*neg_a=*/false, a, /*neg_b=*/false, b,
              /*c_mod=*/(short)0, c, /*reuse_a=*/false, /*reuse_b=*/false);
    }
    acc += c * c;                             // rho += t_l^2 elementwise
  }

  if (lm < NO) {
    #pragma unroll
    for (int r = 0; r < 8; ++r) {
      rho[(size_t)(t0 + r + 8 * half) * NO + lm] = acc[r];
    }
  }
}

// ---------------------------------------------------------------------------
// Species-gathered MLP: h = leaky_relu(rho @ W1[s] + b1[s]); g = h @ W2[s] + b2[s];
// Csn += g.  Thread per atom; all 4 species' weights staged in LDS (~43KB)
// once per block, inner rows read back as float4 (ds_load_b128).
// ---------------------------------------------------------------------------
__global__ void k_mlp(const float* __restrict__ rho,
                      const float* __restrict__ W1,
                      const float* __restrict__ b1,
                      const float* __restrict__ W2,
                      const float* __restrict__ b2,
                      const int* __restrict__ symbols,
                      float* __restrict__ Csn) {
  __shared__ float sW1[NS * NO * HID];   // 2048
  __shared__ float sB1[NS * HID];        // 512
  __shared__ float sW2[NS * HID * NMAX]; // 8192
  __shared__ float sB2[NS * NMAX];       // 64

  for (int i = threadIdx.x; i < NS * NO * HID; i += blockDim.x) sW1[i] = W1[i];
  for (int i = threadIdx.x; i < NS * HID; i += blockDim.x)      sB1[i] = b1[i];
  for (int i = threadIdx.x; i < NS * HID * NMAX; i += blockDim.x) sW2[i] = W2[i];
  for (int i = threadIdx.x; i < NS * NMAX; i += blockDim.x)     sB2[i] = b2[i];
  __syncthreads();

  int t = blockIdx.x * blockDim.x + threadIdx.x;
  if (t >= NTA) return;
  int s = symbols[t];
  float r0 = rho[t * NO + 0];
  float r1 = rho[t * NO + 1];
  float r2 = rho[t * NO + 2];
  float r3 = rho[t * NO + 3];
  const float* w1  = sW1 + s * NO * HID;
  const float* bb1 = sB1 + s * HID;
  const float* w2  = sW2 + s * HID * NMAX;
  const float* bb2 = sB2 + s * NMAX;

  float g[NMAX];
  #pragma unroll
  for (int m = 0; m < NMAX; ++m) g[m] = bb2[m];

  for (int hh = 0; hh < HID; ++hh) {
    float h = bb1[hh];
    h = fmaf(r0, w1[0 * HID + hh], h);
    h = fmaf(r1, w1[1 * HID + hh], h);
    h = fmaf(r2, w1[2 * HID + hh], h);
    h = fmaf(r3, w1[3 * HID + hh], h);
    h = (h > 0.0f) ? h : 0.01f * h;            // leaky_relu(x, 0.01)
    const float4* w2r = (const float4*)(w2 + hh * NMAX);  // 64B-aligned rows
    #pragma unroll
    for (int q = 0; q < 4; ++q) {
      float4 w = w2r[q];
      g[4 * q + 0] = fmaf(h, w.x, g[4 * q + 0]);
      g[4 * q + 1] = fmaf(h, w.y, g[4 * q + 1]);
      g[4 * q + 2] = fmaf(h, w.z, g[4 * q + 2]);
      g[4 * q + 3] = fmaf(h, w.w, g[4 * q + 3]);
    }
  }
  #pragma unroll
  for (int m = 0; m < NMAX; ++m) Csn[t * NMAX + m] += g[m];
}

// ---------------------------------------------------------------------------
// Host launcher
// ---------------------------------------------------------------------------
extern "C" void kernel_launch(void* const* d_in, const int* in_sizes, int n_in,
                              void* d_out, int out_size, void* d_ws, size_t ws_size,
                              hipStream_t stream) {
  const float* disp           = (const float*)d_in[0];
  const float* dist           = (const float*)d_in[1];
  const float* alpha          = (const float*)d_in[2];
  const float* rs             = (const float*)d_in[3];
  const float* species_params = (const float*)d_in[4];
  const float* orbital_params = (const float*)d_in[5];   // 3 x 2 x 16 x 4
  const float* W1             = (const float*)d_in[6];
  const float* b1             = (const float*)d_in[7];
  const float* W2             = (const float*)d_in[8];
  const float* b2             = (const float*)d_in[9];
  const int*   symbols        = (const int*)d_in[10];
  const int*   iidx           = (const int*)d_in[11];
  const int*   jidx           = (const int*)d_in[12];
  const int*   jsym           = (const int*)d_in[13];
  float* out = (float*)d_out;

  // workspace layout (floats): Csn | bnl | rho
  float* Csn = (float*)d_ws;                   // NTA*16
  float* bnl = Csn + (size_t)NTA * NMAX;       // NTA*64
  float* rho = bnl + (size_t)NTA * NO * NMAX;  // NTA*4

  k_init_csn<<<(NTA * NMAX + 255) / 256, 256, 0, stream>>>(species_params, symbols, Csn);

  const int nDensityBlocks = ((NTA / 16) + 7) / 8;   // 8 waves (tiles) per 256-thread block

  for (int p = 0; p < LOOPN + 1; ++p) {
    k_zero<<<(NTA * NO * NMAX + 255) / 256, 256, 0, stream>>>(bnl, NTA * NO * NMAX);
    k_pairs<<<(NN * NMAX + 255) / 256, 256, 0, stream>>>(
        disp, dist, alpha, rs, Csn, iidx, jidx, jsym, bnl);
    float* rout = (p == LOOPN) ? out : rho;
    k_density<<<nDensityBlocks, 256, 0, stream>>>(
        bnl, orbital_params + (size_t)p * LMAX * NMAX * NO, rout);
    if (p < LOOPN) {
      k_mlp<<<(NTA + 255) / 256, 256, 0, stream>>>(rho, W1, b1, W2, b2, symbols, Csn);
    }
  }
}